// DecodePredictions_33767032881981
// MI455X (gfx1250) — compile-verified
//
#include <hip/hip_runtime.h>
#include <hip/hip_bf16.h>
#include <stdint.h>

#define NUM_B   8
#define NUM_N   27621
#define NUM_C   80
#define NPAD    27648          // N padded to multiple of 256 (and 16)
#define MAXPC   100
#define MAXDET  100
#define CONF_T  0.05f
#define IOU_T   0.5f
#define NEGV    (-1.0e9f)
#define NEGH    (-5.0e8f)

typedef __attribute__((ext_vector_type(2))) float v2f;
typedef __attribute__((ext_vector_type(4))) float v4f;
typedef __attribute__((ext_vector_type(8))) float v8f;

__device__ __forceinline__ float sigmoidf_(float x) { return 1.0f / (1.0f + __expf(-x)); }

// ---------------------------------------------------------------- anchors
__device__ __forceinline__ void anchor_of(int n, float& cx, float& cy, float& aw, float& ah)
{
    int lvl, base;
    if      (n < 20736) { lvl = 0; base = 0;     }
    else if (n < 25920) { lvl = 1; base = 20736; }
    else if (n < 27216) { lvl = 2; base = 25920; }
    else if (n < 27540) { lvl = 3; base = 27216; }
    else                { lvl = 4; base = 27540; }
    const int   fws[5]     = {48, 24, 12, 6, 3};
    const float strides[5] = {8.f, 16.f, 32.f, 64.f, 128.f};
    int idx  = n - base;
    int cell = idx / 9, a = idx % 9;
    int fw   = fws[lvl];
    int col  = cell % fw, row = cell / fw;
    float st = strides[lvl];
    cx = (col + 0.5f) * st;
    cy = (row + 0.5f) * st;
    const float sqr[3] = {0.70710678118654752f, 1.0f, 1.41421356237309505f}; // sqrt(ratio)
    const float scl[3] = {1.0f, 1.2599210498948732f, 1.5874010519681994f};   // 2^(k/3)
    int ri = a / 3, si = a % 3;
    float s4  = 4.0f * st;                 // sqrt(area), area = (4*stride)^2
    float ahh = s4 / sqr[ri];              // sqrt(area / r)
    float aww = s4 * sqr[ri];              // area / ah
    aw = aww * scl[si];
    ah = ahh * scl[si];
}

// ---------------------------------------------------------------- box decode
__global__ void dp_decode_boxes(const float* __restrict__ pred, float* __restrict__ boxes)
{
    int t = blockIdx.x * blockDim.x + threadIdx.x;
    if (t >= NUM_B * NUM_N) return;
    int b = t / NUM_N, n = t % NUM_N;
    const float* p = pred + ((size_t)b * NUM_N + n) * (4 + NUM_C);
    float cx, cy, aw, ah;
    anchor_of(n, cx, cy, aw, ah);
    float x = p[0] * 0.1f * aw + cx;
    float y = p[1] * 0.1f * ah + cy;
    float w = expf(p[2] * 0.2f) * aw;
    float h = expf(p[3] * 0.2f) * ah;
    float4 bb;
    bb.x = x - 0.5f * w; bb.y = y - 0.5f * h;
    bb.z = x + 0.5f * w; bb.w = y + 0.5f * h;
    ((float4*)boxes)[t] = bb;
}

// ------------------------------------------- sigmoid + transpose via WMMA
// scoresT[b][c][n] = sigmoid(pred[b][n][4+c]); WMMA f32 16x16x4 used as an
// exact tile copier into the D layout: D = sum_q A_q x E_q, E_q one-hot.
#if __has_builtin(__builtin_amdgcn_wmma_f32_16x16x4_f32)
#define DP_HAVE_WMMA_F32 1
#endif

__global__ void dp_sig_transpose(const float* __restrict__ pred, float* __restrict__ scoresT)
{
#ifdef DP_HAVE_WMMA_F32
    int lane = threadIdx.x & 31;
    int wave = (blockIdx.x * blockDim.x + threadIdx.x) >> 5;
    const int tilesPerImg = (NPAD / 16) * 5;   // 1728 * 5
    int b  = wave / tilesPerImg;
    int r  = wave % tilesPerImg;
    int tn = r / 5, tc = r % 5;
    int n0 = tn * 16, c0 = tc * 16;
    int m  = lane & 15, hi = lane >> 4;
    int row = n0 + m; if (row > NUM_N - 1) row = NUM_N - 1;   // clamp; lands in masked pad region
    const float* src = pred + ((size_t)b * NUM_N + row) * (4 + NUM_C) + 4 + c0 + 2 * hi;
    v8f acc = {};
#pragma unroll
    for (int q = 0; q < 4; ++q) {
        // A tile: lane holds T[m][4q + 2*hi + {0,1}]
        v2f A;
        A.x = sigmoidf_(src[4 * q + 0]);
        A.y = sigmoidf_(src[4 * q + 1]);
        // Selector E_q: B[k][n] = (n == 4q + k); lane holds B[2*hi + v][m]
        int k0 = 4 * q + 2 * hi;
        v2f Bm;
        Bm.x = (m == k0)     ? 1.0f : 0.0f;
        Bm.y = (m == k0 + 1) ? 1.0f : 0.0f;
        acc = __builtin_amdgcn_wmma_f32_16x16x4_f32(false, A, false, Bm,
                                                    (short)0, acc, false, false);
    }
    // D layout: lane l owns column (l&15), rows 8*(l>>4) .. +7 -> coalesced 32B store
    int col = c0 + m;
    float* dst = scoresT + ((size_t)b * NUM_C + col) * NPAD + n0 + 8 * hi;
    v4f lo = {acc[0], acc[1], acc[2], acc[3]};
    v4f hv = {acc[4], acc[5], acc[6], acc[7]};
    *(v4f*)dst       = lo;
    *(v4f*)(dst + 4) = hv;
#else
    // scalar fallback (grid-stride)
    const size_t total = (size_t)NUM_B * NUM_C * NUM_N;
    for (size_t t = (size_t)blockIdx.x * blockDim.x + threadIdx.x; t < total;
         t += (size_t)gridDim.x * blockDim.x) {
        size_t bc = t / NUM_N;
        int n = (int)(t % NUM_N);
        int b = (int)(bc / NUM_C), c = (int)(bc % NUM_C);
        scoresT[(bc)*NPAD + n] = sigmoidf_(pred[((size_t)b * NUM_N + n) * (4 + NUM_C) + 4 + c]);
    }
#endif
}

// ------------------------------------------------------- per-(b,c) greedy NMS
// One workgroup per (image, class). Scores live in LDS (async-staged),
// boxes streamed from the 192MB L2. Fused suppress + next-argmax pass.
__global__ void dp_nms(const float* __restrict__ scoresT, const float* __restrict__ boxes,
                       float* __restrict__ candVal, int* __restrict__ candIdx)
{
    extern __shared__ float smem[];           // [NPAD] scores, +32 floats scratch
    const int NT  = NPAD;
    const int tid = threadIdx.x;
    const int b   = blockIdx.x / NUM_C;
    const int c   = blockIdx.x % NUM_C;
    const float* srow = scoresT + ((size_t)b * NUM_C + c) * NPAD;

    // ---- async global -> LDS staging (b128 per lane), gfx1250 path
    for (int i = tid * 4; i < NT; i += blockDim.x * 4) {
        unsigned lds = (unsigned)(i * 4);
        unsigned long long g = (unsigned long long)(const void*)(srow + i);
        asm volatile("global_load_async_to_lds_b128 %0, %1, off"
                     :: "v"(lds), "v"(g) : "memory");
    }
    asm volatile("s_wait_asynccnt 0" ::: "memory");
    __syncthreads();

    // ---- threshold in place: s0 = score > CONF ? score : NEG ; pads -> NEG
    for (int i = tid; i < NT; i += blockDim.x) {
        float v = smem[i];
        smem[i] = (i < NUM_N && v > CONF_T) ? v : NEGV;
    }
    __syncthreads();

    // ---- initial local argmax (per-thread strided; '>' keeps lowest index)
    float bv = -3.4e38f; int bi = 0;
    for (int i = tid; i < NT; i += blockDim.x) {
        float v = smem[i];
        if (v > bv) { bv = v; bi = i; }
    }

    const float4* bxs = (const float4*)boxes + (size_t)b * NUM_N;
    float* cV = candVal + ((size_t)b * NUM_C + c) * MAXPC;
    int*   cI = candIdx + ((size_t)b * NUM_C + c) * MAXPC;
    float* pv  = smem + NT;            // 8 wave partial values
    int*   pi  = (int*)(smem + NT + 8);
    float* win = smem + NT + 16;       // [0..3] box, [4] unused
    int*   wI  = (int*)(smem + NT + 21); // [0] idx, [1] valid

    const int wid = tid >> 5, lane = tid & 31;

    for (int mIt = 0; mIt < MAXPC; ++mIt) {
        // wave32 butterfly reduce (max value, lowest index tie-break)
        for (int off = 16; off > 0; off >>= 1) {
            float ov = __shfl_xor(bv, off, 32);
            int   oi = __shfl_xor(bi, off, 32);
            if (ov > bv || (ov == bv && oi < bi)) { bv = ov; bi = oi; }
        }
        if (lane == 0) { pv[wid] = bv; pi[wid] = bi; }
        __syncthreads();
        if (tid == 0) {
            float wv = pv[0]; int wi = pi[0];
            for (int w = 1; w < 8; ++w) {
                float ov = pv[w]; int oi = pi[w];
                if (ov > wv || (ov == wv && oi < wi)) { wv = ov; wi = oi; }
            }
            int valid = (wv > NEGH) ? 1 : 0;
            cV[mIt] = valid ? wv : NEGV;
            cI[mIt] = wi;
            if (!valid)
                for (int mm = mIt + 1; mm < MAXPC; ++mm) { cV[mm] = NEGV; cI[mm] = wi; }
            float4 wb = bxs[(wi >= 0 && wi < NUM_N) ? wi : 0];
            win[0] = wb.x; win[1] = wb.y; win[2] = wb.z; win[3] = wb.w;
            wI[0] = wi; wI[1] = valid;
        }
        __syncthreads();
        if (!wI[1]) break;                         // uniform; remaining rounds all invalid

        float wx1 = win[0], wy1 = win[1], wx2 = win[2], wy2 = win[3];
        float aW  = (wx2 - wx1) * (wy2 - wy1);

        // fused suppression + next local argmax (each thread owns its stride)
        bv = -3.4e38f; bi = 0;
        for (int i = tid; i < NT; i += blockDim.x) {
            float v = smem[i];
            if (v > NEGH) {                        // active (pads are NEG -> i < NUM_N)
                float4 bb = bxs[i];
                float lx = fmaxf(wx1, bb.x), ly = fmaxf(wy1, bb.y);
                float rx = fminf(wx2, bb.z), ry = fminf(wy2, bb.w);
                float iw = fmaxf(rx - lx, 0.f), ih = fmaxf(ry - ly, 0.f);
                float inter = iw * ih;
                float aB = (bb.z - bb.x) * (bb.w - bb.y);
                float iou = inter / (aW + aB - inter + 1e-8f);
                if (iou > IOU_T) { v = NEGV; smem[i] = NEGV; }
            }
            if (v > bv) { bv = v; bi = i; }
        }
    }
}

// ------------------------------------------------ per-image top-100 + assembly
__global__ void dp_topk(const float* __restrict__ boxes, float* __restrict__ candVal,
                        const int* __restrict__ candIdx, float* __restrict__ out)
{
    __shared__ float pv[8];
    __shared__ int   pi[8];
    const int b = blockIdx.x, tid = threadIdx.x;
    float*       cv   = candVal + (size_t)b * NUM_C * MAXPC;
    const int*   ci   = candIdx + (size_t)b * NUM_C * MAXPC;
    const float4* bxs = (const float4*)boxes + (size_t)b * NUM_N;
    float* dense = out + (size_t)b * MAXDET * 6;
    const int wid = tid >> 5, lane = tid & 31;
    const int TOT = NUM_C * MAXPC;
    int vcount = 0;   // meaningful in tid 0 only

    for (int j = 0; j < MAXDET; ++j) {
        float bv = -3.4e38f; int bi = 0;
        for (int t = tid; t < TOT; t += blockDim.x) {
            float v = cv[t];
            if (v > bv) { bv = v; bi = t; }        // strided ascending: keeps lowest t
        }
        for (int off = 16; off > 0; off >>= 1) {
            float ov = __shfl_xor(bv, off, 32);
            int   oi = __shfl_xor(bi, off, 32);
            if (ov > bv || (ov == bv && oi < bi)) { bv = ov; bi = oi; }
        }
        if (lane == 0) { pv[wid] = bv; pi[wid] = bi; }
        __syncthreads();
        if (tid == 0) {
            float wv = pv[0]; int wt = pi[0];
            for (int w = 1; w < 8; ++w) {
                float ov = pv[w]; int oi = pi[w];
                if (ov > wv || (ov == wv && oi < wt)) { wv = ov; wt = oi; }
            }
            int valid = (wv > NEGH) ? 1 : 0;
            float4 bb = {0.f, 0.f, 0.f, 0.f};
            float clsf = 0.f, sc = 0.f;
            if (valid) {
                int bix = ci[wt];
                bb   = bxs[(bix >= 0 && bix < NUM_N) ? bix : 0];
                clsf = (float)(wt / MAXPC);
                sc   = wv;
            }
            dense[j * 6 + 0] = bb.x; dense[j * 6 + 1] = bb.y;
            dense[j * 6 + 2] = bb.z; dense[j * 6 + 3] = bb.w;
            dense[j * 6 + 4] = clsf; dense[j * 6 + 5] = sc;
            vcount += valid;
            cv[wt] = -3.4e38f;                     // exclude from further rounds
        }
        __syncthreads();
    }
    if (tid == 0) out[(size_t)NUM_B * MAXDET * 6 + b] = (float)vcount;
}

// ---------------------------------------------------------------- launcher
extern "C" void kernel_launch(void* const* d_in, const int* in_sizes, int n_in,
                              void* d_out, int out_size, void* d_ws, size_t ws_size,
                              hipStream_t stream)
{
    (void)in_sizes; (void)n_in; (void)out_size; (void)ws_size;
    const float* pred = (const float*)d_in[1];   // d_in[0] = images (only shape used)
    float* wsF = (float*)d_ws;
    float* boxes   = wsF;                                           // B*N*4      f32
    float* scoresT = boxes   + (size_t)NUM_B * NUM_N * 4;           // B*C*NPAD   f32
    float* candVal = scoresT + (size_t)NUM_B * NUM_C * NPAD;        // B*C*100    f32
    int*   candIdx = (int*)(candVal + (size_t)NUM_B * NUM_C * MAXPC); // B*C*100  i32
    float* out = (float*)d_out;

    dim3 blk(256);
    dp_decode_boxes<<<(NUM_B * NUM_N + 255) / 256, blk, 0, stream>>>(pred, boxes);
    dp_sig_transpose<<<(NUM_B * (NPAD / 16) * 5) / 8, blk, 0, stream>>>(pred, scoresT);

    size_t smemB = (size_t)(NPAD + 32) * sizeof(float);   // ~110.7 KB of the 320 KB WGP LDS
    hipFuncSetAttribute((const void*)dp_nms,
                        hipFuncAttributeMaxDynamicSharedMemorySize, (int)smemB);
    dp_nms<<<NUM_B * NUM_C, blk, smemB, stream>>>(scoresT, boxes, candVal, candIdx);
    dp_topk<<<NUM_B, blk, 0, stream>>>(boxes, candVal, candIdx, out);
}